// PatchSelection_25795573580084
// MI455X (gfx1250) — compile-verified
//
#include <hip/hip_runtime.h>
#include <hip/hip_bf16.h>

#define B_DIM      64
#define L_DIM      4096
#define D_DIM      384
#define LEN_KEEP   162
#define ROWS_OUT   (LEN_KEEP + 1)          // 163
#define ROWS_TOTAL (B_DIM * L_DIM)         // 262144

typedef __attribute__((ext_vector_type(2))) float v2f;
typedef __attribute__((ext_vector_type(8))) float v8f;

// ---------------------------------------------------------------------------
// Kernel 1: row means via V_WMMA_F32_16X16X4_F32.
// One wave32 per 16 rows. A-fragment layout (16x4 f32): lanes 0-15 hold
// {K=0,K=1} of row M=lane, lanes 16-31 hold {K=2,K=3} of row M=lane-16.
// B is the constant 1/384 in every slot, so the K-slot assignment of x's
// columns is irrelevant (sum is permutation-invariant) -- we just need each
// of the 384 columns to appear in exactly one A slot across the 96 WMMAs.
// Per step j: lanes 0-15 load cols [8j,8j+4) of their row, lanes 16-31 load
// cols [8j+4,8j+8) of theirs; WMMA#1 consumes (f.x,f.y), WMMA#2 (f.z,f.w).
// D layout: acc VGPR v on lane 0  -> mean(row0+v)   (N=0 column),
//           acc VGPR v on lane 16 -> mean(row0+8+v).
// ---------------------------------------------------------------------------
__global__ __launch_bounds__(32)
void ps_mean_wmma_kernel(const float* __restrict__ x, float* __restrict__ means) {
    const int row0 = blockIdx.x * 16;
    const int lane = threadIdx.x;          // 0..31, EXEC all ones (WMMA req.)
    const int r    = lane & 15;
    const int half = lane >> 4;            // 0: K=0,1 slots   1: K=2,3 slots
    const float* rowp = x + (size_t)(row0 + r) * D_DIM;

    const float c = 1.0f / (float)D_DIM;
    v2f bfrag; bfrag[0] = c; bfrag[1] = c;
    v8f acc = {};

    #pragma unroll 4
    for (int j = 0; j < D_DIM / 8; ++j) {
        const float4 f = *(const float4*)(rowp + j * 8 + half * 4);
        v2f a1; a1[0] = f.x; a1[1] = f.y;
        v2f a2; a2[0] = f.z; a2[1] = f.w;
        acc = __builtin_amdgcn_wmma_f32_16x16x4_f32(
                  false, a1, false, bfrag, (short)0, acc, false, false);
        acc = __builtin_amdgcn_wmma_f32_16x16x4_f32(
                  false, a2, false, bfrag, (short)0, acc, false, false);
    }

    if (r == 0) {                          // lane 0 -> rows 0..7, lane 16 -> 8..15
        float* out = means + row0 + half * 8;
        #pragma unroll
        for (int v = 0; v < 8; ++v) out[v] = acc[v];
    }
}

// ---------------------------------------------------------------------------
// Kernel 2: per-batch top-162 (descending, ties -> smaller index, matching
// jax.lax.top_k). One block per batch; scores staged in LDS; 162 rounds of
// argmax + knockout. 42M comparisons total -- negligible vs the
// streaming pass.
// ---------------------------------------------------------------------------
__global__ __launch_bounds__(256)
void ps_topk_kernel(const float* __restrict__ means, int* __restrict__ sel) {
    __shared__ float s_val[L_DIM];
    __shared__ float r_val[256];
    __shared__ int   r_idx[256];

    const int b = blockIdx.x;
    const int t = threadIdx.x;
    const float* m = means + (size_t)b * L_DIM;

    for (int i = t; i < L_DIM; i += 256) s_val[i] = m[i];
    __syncthreads();

    for (int k = 0; k < LEN_KEEP; ++k) {
        float best = -__builtin_inff();
        int   bi   = 0x7FFFFFFF;
        for (int i = t; i < L_DIM; i += 256) {
            const float v = s_val[i];
            if (v > best || (v == best && i < bi)) { best = v; bi = i; }
        }
        r_val[t] = best; r_idx[t] = bi;
        __syncthreads();
        for (int s = 128; s > 0; s >>= 1) {
            if (t < s) {
                const float v  = r_val[t + s];
                const int   i2 = r_idx[t + s];
                if (v > r_val[t] || (v == r_val[t] && i2 < r_idx[t])) {
                    r_val[t] = v; r_idx[t] = i2;
                }
            }
            __syncthreads();
        }
        if (t == 0) {
            const int w = r_idx[0];
            sel[(size_t)b * LEN_KEEP + k] = w;
            s_val[w] = -__builtin_inff();            // knock out winner
        }
        __syncthreads();
    }
}

// ---------------------------------------------------------------------------
// Kernel 3: build (B, 163, 384) output: row 0 = token[b], rows 1..162 =
// gathered x rows. float4 copies, 96 lanes per row. Also emits the selected
// indices (exact in f32) into the tail of d_out.
// ---------------------------------------------------------------------------
__global__ __launch_bounds__(96)
void ps_gather_kernel(const float* __restrict__ x,
                      const float* __restrict__ token,
                      const int*   __restrict__ sel,
                      float*       __restrict__ out_x,
                      float*       __restrict__ out_idx) {
    const int b = blockIdx.y;              // 0..63
    const int r = blockIdx.x;              // 0..162
    const int t = threadIdx.x;             // 0..95 (float4 lanes)

    float4 v;
    if (r == 0) {
        v = *(const float4*)(token + (size_t)b * D_DIM + t * 4);
    } else {
        const int src = sel[(size_t)b * LEN_KEEP + (r - 1)];
        v = *(const float4*)(x + ((size_t)b * L_DIM + src) * D_DIM + t * 4);
        if (t == 0) out_idx[(size_t)b * LEN_KEEP + (r - 1)] = (float)src;
    }
    *(float4*)(out_x + ((size_t)b * ROWS_OUT + r) * D_DIM + t * 4) = v;
}

// ---------------------------------------------------------------------------
extern "C" void kernel_launch(void* const* d_in, const int* in_sizes, int n_in,
                              void* d_out, int out_size, void* d_ws, size_t ws_size,
                              hipStream_t stream) {
    const float* x     = (const float*)d_in[0];   // (64, 4096, 384) f32
    const float* token = (const float*)d_in[1];   // (64, 384) f32

    float* out_x   = (float*)d_out;                               // 64*163*384
    float* out_idx = (float*)d_out + (size_t)B_DIM * ROWS_OUT * D_DIM;

    float* means = (float*)d_ws;                                  // 1 MiB
    int*   sel   = (int*)((char*)d_ws + (size_t)ROWS_TOTAL * sizeof(float));

    ps_mean_wmma_kernel<<<ROWS_TOTAL / 16, 32, 0, stream>>>(x, means);
    ps_topk_kernel<<<B_DIM, 256, 0, stream>>>(means, sel);
    dim3 g(ROWS_OUT, B_DIM);
    ps_gather_kernel<<<g, 96, 0, stream>>>(x, token, sel, out_x, out_idx);
}